// GraphNetworkBlock_79233556677177
// MI455X (gfx1250) — compile-verified
//
#include <hip/hip_runtime.h>

typedef __attribute__((ext_vector_type(2))) float v2f;
typedef __attribute__((ext_vector_type(8))) float v8f;

#define BS 8
#define NN 256
#define GD 64
#define ND 64
#define ED 32

// ---------------------------------------------------------------------------
// K0: Vj = V @ Wa_vj, Vi = V @ Wa_vi, Ub = u @ Wa_u + b_A
// gid over BS*NN*ED = 65536
// ---------------------------------------------------------------------------
__global__ void k0_precompute(const float* __restrict__ u,
                              const float* __restrict__ V,
                              const float* __restrict__ W_A,
                              const float* __restrict__ b_A,
                              float* __restrict__ Vj,
                              float* __restrict__ Vi,
                              float* __restrict__ Ub) {
  int gid = blockIdx.x * blockDim.x + threadIdx.x;   // 0..65535
  int e = gid & (ED - 1);
  int n = (gid >> 5) & (NN - 1);
  int b = gid >> 13;
  const float* vrow = V + (b * NN + n) * ND;
  float aj = 0.f, ai = 0.f;
  for (int k = 0; k < ND; ++k) {
    float v = vrow[k];
    aj += v * W_A[(ED + k) * ED + e];           // Wa_vj rows [32..96)
    ai += v * W_A[(ED + ND + k) * ED + e];      // Wa_vi rows [96..160)
  }
  Vj[gid] = aj;
  Vi[gid] = ai;
  if (gid < BS * ED) {                          // 256 threads also build Ub
    int bb = gid >> 5;
    int ee = gid & (ED - 1);
    float a = b_A[ee];
    for (int k = 0; k < GD; ++k)
      a += u[bb * GD + k] * W_A[(ED + 2 * ND + k) * ED + ee];  // Wa_u rows [160..224)
    Ub[gid] = a;
  }
}

// ---------------------------------------------------------------------------
// K1: main edge kernel.
//   A'[b,i,j,:] = relu(A[b,i,j,:] @ Wa_e + Vj[b,j] + Vi[b,i] + Ub[b])
// via V_WMMA_F32_16X16X4_F32:  M = 16 consecutive j, N = 16-wide out tile,
// K accumulated 4 at a time over ED=32.
// grid = (16 j-tiles, 8 i-chunks, 8 batches), block = 128 (4 waves).
// Wave w handles i = ic*32 + 4*t + w, t = 0..7.
//
// f32 WMMA fragment layouts (ISA 7.12.2):
//   A 16x4 : lane L -> M = L%16 ; VGPR v -> K = v + 2*(L/16)
//   B 4x16 : lane L -> N = L%16 ; VGPR v -> K = v + 2*(L/16)
//   C/D    : lane L, VGPR r -> (M = r + 8*(L/16), N = L%16)
// ---------------------------------------------------------------------------
__global__ void __launch_bounds__(128)
k1_edge(const float* __restrict__ A,
        const float* __restrict__ W_A,
        const float* __restrict__ Vj,
        const float* __restrict__ Vi,
        const float* __restrict__ Ub,
        float* __restrict__ Aout,
        float* __restrict__ part) {
  const int lane   = threadIdx.x & 31;
  const int w      = threadIdx.x >> 5;
  const int jt     = blockIdx.x;     // 0..15
  const int ic     = blockIdx.y;     // 0..7
  const int b      = blockIdx.z;     // 0..7
  const int j0     = jt * 16;
  const int lhalf  = lane >> 4;      // 0/1
  const int lm     = lane & 15;

  // --- B fragments: Wa_e[K][N] = W_A rows [0..32). Held for both N tiles, all 8 K-blocks.
  v2f bfrag[2][8];
#pragma unroll
  for (int nt = 0; nt < 2; ++nt) {
#pragma unroll
    for (int kb = 0; kb < 8; ++kb) {
      int kk  = 4 * kb + 2 * lhalf;
      int col = nt * 16 + lm;
      v2f t;
      t[0] = W_A[kk * ED + col];
      t[1] = W_A[(kk + 1) * ED + col];
      bfrag[nt][kb] = t;
    }
  }

  // Per-lane broadcast operands for the epilogue (depend only on N).
  const float ub0 = Ub[b * ED + lm];
  const float ub1 = Ub[b * ED + 16 + lm];

  float sum[2][8];
#pragma unroll
  for (int nt = 0; nt < 2; ++nt)
#pragma unroll
    for (int r = 0; r < 8; ++r) sum[nt][r] = 0.f;

  for (int t = 0; t < 8; ++t) {
    const int i = ic * 32 + t * 4 + w;
    const float* arow =
        A + (((size_t)(b * NN) + i) * NN + (j0 + lm)) * ED + 2 * lhalf;

    if (t < 7)  // prefetch next i's row segment (i advances by 4)
      __builtin_prefetch(arow + (size_t)4 * NN * ED, 0, 1);

    v2f af[8];
#pragma unroll
    for (int kb = 0; kb < 8; ++kb)
      af[kb] = *reinterpret_cast<const v2f*>(arow + 4 * kb);

    v8f c0 = {};
    v8f c1 = {};
#pragma unroll
    for (int kb = 0; kb < 8; ++kb) {
      c0 = __builtin_amdgcn_wmma_f32_16x16x4_f32(false, af[kb],
                                                 false, bfrag[0][kb],
                                                 (short)0, c0, false, false);
      c1 = __builtin_amdgcn_wmma_f32_16x16x4_f32(false, af[kb],
                                                 false, bfrag[1][kb],
                                                 (short)0, c1, false, false);
    }

    const float vi0 = Vi[(b * NN + i) * ED + lm];
    const float vi1 = Vi[(b * NN + i) * ED + 16 + lm];
    float* orow = Aout + (((size_t)(b * NN) + i) * NN + j0) * ED;

#pragma unroll
    for (int r = 0; r < 8; ++r) {
      const int jj = r + 8 * lhalf;                     // local row (j - j0)
      const float vj0 = Vj[(b * NN + j0 + jj) * ED + lm];
      const float vj1 = Vj[(b * NN + j0 + jj) * ED + 16 + lm];
      float x0 = c0[r] + vj0 + vi0 + ub0; x0 = x0 > 0.f ? x0 : 0.f;
      float x1 = c1[r] + vj1 + vi1 + ub1; x1 = x1 > 0.f ? x1 : 0.f;
      orow[jj * ED + lm]      = x0;
      orow[jj * ED + 16 + lm] = x1;
      sum[0][r] += x0;
      sum[1][r] += x1;
    }
  }

  // --- block-level reduction of the i-partial sums (per (jj,e)) into part[ic]
  __shared__ float lred[4][16][ED];
#pragma unroll
  for (int nt = 0; nt < 2; ++nt)
#pragma unroll
    for (int r = 0; r < 8; ++r) {
      const int jj = r + 8 * lhalf;
      lred[w][jj][nt * 16 + lm] = sum[nt][r];
    }
  __syncthreads();
  for (int f = threadIdx.x; f < 16 * ED; f += 128) {
    const int jj = f >> 5;
    const int e  = f & (ED - 1);
    const float s = lred[0][jj][e] + lred[1][jj][e] + lred[2][jj][e] + lred[3][jj][e];
    part[(size_t)ic * (BS * NN * ED) + (b * NN + j0 + jj) * ED + e] = s;
  }
}

// ---------------------------------------------------------------------------
// K1b: colsum[b,j,e] = sum over 8 i-chunk partials  ( = 256 * A_agg_by_node )
// ---------------------------------------------------------------------------
__global__ void k1b_colsum(const float* __restrict__ part,
                           float* __restrict__ colsum) {
  int gid = blockIdx.x * blockDim.x + threadIdx.x;  // 0..65535
  float s = 0.f;
#pragma unroll
  for (int c = 0; c < 8; ++c) s += part[(size_t)c * (BS * NN * ED) + gid];
  colsum[gid] = s;
}

// ---------------------------------------------------------------------------
// K2: V' = relu([colsum/256, V, u] @ W_v + b_v) ; one thread per output elem
// ---------------------------------------------------------------------------
__global__ void k2_vprime(const float* __restrict__ colsum,
                          const float* __restrict__ V,
                          const float* __restrict__ u,
                          const float* __restrict__ W_v,
                          const float* __restrict__ b_v,
                          float* __restrict__ Vout) {
  int gid = blockIdx.x * blockDim.x + threadIdx.x;  // BS*NN*ND
  int o = gid & (ND - 1);
  int n = (gid >> 6) & (NN - 1);
  int b = gid >> 14;
  float acc = b_v[o];
  const float inv = 1.f / 256.f;
  const float* arow = colsum + (b * NN + n) * ED;
  for (int e = 0; e < ED; ++e) acc += arow[e] * inv * W_v[e * ND + o];
  const float* vrow = V + (b * NN + n) * ND;
  for (int k = 0; k < ND; ++k) acc += vrow[k] * W_v[(ED + k) * ND + o];
  const float* urow = u + b * GD;
  for (int k = 0; k < GD; ++k) acc += urow[k] * W_v[(ED + ND + k) * ND + o];
  Vout[gid] = acc > 0.f ? acc : 0.f;
}

// ---------------------------------------------------------------------------
// K3: u' = relu([A_agg, v_agg, u] @ W_u + b_u), one block per batch
// ---------------------------------------------------------------------------
__global__ void k3_uprime(const float* __restrict__ colsum,
                          const float* __restrict__ Vprime,
                          const float* __restrict__ u,
                          const float* __restrict__ W_u,
                          const float* __restrict__ b_u,
                          float* __restrict__ uout) {
  const int b = blockIdx.x;
  const int tid = threadIdx.x;  // 256 threads
  __shared__ float xu[ED + ND + GD];  // 160
  if (tid < ED) {
    float s = 0.f;
    for (int j = 0; j < NN; ++j) s += colsum[(b * NN + j) * ED + tid];
    xu[tid] = s * (1.f / (256.f * 256.f));       // mean over (i,j)
  } else if (tid < ED + ND) {
    const int k = tid - ED;
    float s = 0.f;
    for (int n = 0; n < NN; ++n) s += Vprime[(b * NN + n) * ND + k];
    xu[tid] = s * (1.f / 256.f);                  // mean over nodes
  } else if (tid < ED + ND + GD) {
    xu[tid] = u[b * GD + (tid - ED - ND)];
  }
  __syncthreads();
  if (tid < GD) {
    float acc = b_u[tid];
    for (int k = 0; k < ED + ND + GD; ++k) acc += xu[k] * W_u[k * GD + tid];
    uout[b * GD + tid] = acc > 0.f ? acc : 0.f;
  }
}

// ---------------------------------------------------------------------------
extern "C" void kernel_launch(void* const* d_in, const int* in_sizes, int n_in,
                              void* d_out, int out_size, void* d_ws, size_t ws_size,
                              hipStream_t stream) {
  const float* u   = (const float*)d_in[0];
  const float* V   = (const float*)d_in[1];
  const float* A   = (const float*)d_in[2];
  const float* W_A = (const float*)d_in[3];
  const float* b_A = (const float*)d_in[4];
  const float* W_v = (const float*)d_in[5];
  const float* b_v = (const float*)d_in[6];
  const float* W_u = (const float*)d_in[7];
  const float* b_u = (const float*)d_in[8];

  float* out   = (float*)d_out;
  float* u_out = out;                                   // (8,64)
  float* V_out = out + BS * GD;                         // (8,256,64)
  float* A_out = out + BS * GD + BS * NN * ND;          // (8,256,256,32)

  float* ws     = (float*)d_ws;
  float* Vj     = ws;                                   // 65536
  float* Vi     = Vj + BS * NN * ED;                    // 65536
  float* Ub     = Vi + BS * NN * ED;                    // 256
  float* colsum = Ub + BS * ED;                         // 65536
  float* part   = colsum + BS * NN * ED;                // 8*65536

  k0_precompute<<<dim3((BS * NN * ED) / 256), 256, 0, stream>>>(u, V, W_A, b_A, Vj, Vi, Ub);
  k1_edge<<<dim3(16, 8, 8), 128, 0, stream>>>(A, W_A, Vj, Vi, Ub, A_out, part);
  k1b_colsum<<<dim3((BS * NN * ED) / 256), 256, 0, stream>>>(part, colsum);
  k2_vprime<<<dim3((BS * NN * ND) / 256), 256, 0, stream>>>(colsum, V, u, W_v, b_v, V_out);
  k3_uprime<<<dim3(BS), 256, 0, stream>>>(colsum, V_out, u, W_u, b_u, u_out);
}